// HeavyEncoderLayer_74388833566991
// MI455X (gfx1250) — compile-verified
//
#include <hip/hip_runtime.h>
#include <cstdint>
#include <cstddef>

#define N_NODES 200000
#define N_EDGES 3200000
#define HSEG    100000

#define AS1 __attribute__((address_space(1)))
#define AS3 __attribute__((address_space(3)))

typedef int v2i __attribute__((vector_size(8)));   // 2 x i32, matches b64 builtin

// CDNA5 async global->LDS path (gfx1250). Guarded so host pass / older clang
// falls back to plain LDS staging.
#if defined(__HIP_DEVICE_COMPILE__) && \
    __has_builtin(__builtin_amdgcn_global_load_async_to_lds_b32) && \
    __has_builtin(__builtin_amdgcn_global_load_async_to_lds_b64)
#define HAS_ASYNC 1
#else
#define HAS_ASYNC 0
#endif

struct Tile {
    int   src[64];
    int   dst[64];
    float ea[256];   // 64 edges x 4 attrs, 8B-aligned for b64 async loads
};

__device__ __forceinline__ void afadd(float* p, float v) {
    // hardware global_atomic_add_f32 (no CAS loop)
    unsafeAtomicAdd(p, v);
}

__device__ __forceinline__ void stage_tile(Tile* t,
                                           const int* __restrict__ eidx,
                                           const float* __restrict__ ea,
                                           long e0) {
    const int tid = threadIdx.x;
#if HAS_ASYNC
    if (tid < 64) {
        __builtin_amdgcn_global_load_async_to_lds_b32(
            (AS1 int*)(eidx + e0 + tid),
            (AS3 int*)(&t->src[tid]), 0, 0);
    } else if (tid < 128) {
        const int i = tid - 64;
        __builtin_amdgcn_global_load_async_to_lds_b32(
            (AS1 int*)(eidx + (long)N_EDGES + e0 + i),
            (AS3 int*)(&t->dst[i]), 0, 0);
    } else {
        const int i = tid - 128;                 // 0..127, 2 floats each
        __builtin_amdgcn_global_load_async_to_lds_b64(
            (AS1 v2i*)(ea + e0 * 4 + (long)i * 2),
            (AS3 v2i*)(&t->ea[i * 2]), 0, 0);
    }
#else
    if (tid < 64) {
        t->src[tid] = eidx[e0 + tid];
    } else if (tid < 128) {
        t->dst[tid - 64] = eidx[(long)N_EDGES + e0 + (tid - 64)];
    } else {
        const int i = tid - 128;
        t->ea[2 * i + 0] = ea[e0 * 4 + 2 * i + 0];
        t->ea[2 * i + 1] = ea[e0 * 4 + 2 * i + 1];
    }
#endif
}

__device__ __forceinline__ void pipeline_wait() {
#if HAS_ASYNC
#if __has_builtin(__builtin_amdgcn_s_wait_asynccnt)
    __builtin_amdgcn_s_wait_asynccnt(0);
#else
    asm volatile("s_wait_asynccnt 0" ::: "memory");
#endif
#endif
    __syncthreads();
}

// ---------------------------------------------------------------------------
// Kernel 1: per-edge messages, scatter-add into node accumulators (= d_out).
// 16 lanes per edge (lane u = channel), 64-edge tiles double-buffered in LDS
// via CDNA5 async-to-LDS loads. 8 tiles (512 edges) per block.
// ---------------------------------------------------------------------------
__global__ __launch_bounds__(256) void edge_msg_kernel(
    const float* __restrict__ x, const float* __restrict__ edge_attr,
    const float* __restrict__ w_msg, const int* __restrict__ edge_index,
    float* __restrict__ acc) {
    __shared__ __align__(16) Tile tiles[2];
    const int tid = threadIdx.x;
    const int u = tid & 15;
    const long base = (long)blockIdx.x * 512;

    const float w0 = w_msg[u];
    const float w1 = w_msg[16 + u];
    const float w2 = w_msg[32 + u];
    const float w3 = w_msg[48 + u];
    const float w4 = w_msg[64 + u];

    stage_tile(&tiles[0], edge_index, edge_attr, base);

    for (int t = 0; t < 8; ++t) {
        pipeline_wait();  // tile t resident in LDS for all waves
        if (t + 1 < 8)
            stage_tile(&tiles[(t + 1) & 1], edge_index, edge_attr,
                       base + (long)(t + 1) * 64);
        const Tile* cur = &tiles[t & 1];
#pragma unroll
        for (int sub = 0; sub < 4; ++sub) {
            const int eL = sub * 16 + (tid >> 4);
            const int s = cur->src[eL];
            const int d = cur->dst[eL];
            const float es  = cur->ea[eL * 4 + 0];
            const float ev0 = cur->ea[eL * 4 + 1];
            const float ev1 = cur->ea[eL * 4 + 2];
            const float ev2 = cur->ea[eL * 4 + 3];

            const float* xr = x + (size_t)s * 64;
            const float xs = xr[u];
            const float a0 = xr[16 + 3 * u + 0];
            const float a1 = xr[16 + 3 * u + 1];
            const float a2 = xr[16 + 3 * u + 2];

            const float ms = w0 * xs * es + w1 * (a0 * ev0 + a1 * ev1 + a2 * ev2);
            const float c0 = a1 * ev2 - a2 * ev1;
            const float c1 = a2 * ev0 - a0 * ev2;
            const float c2 = a0 * ev1 - a1 * ev0;
            const float m0 = w2 * a0 * es + w3 * xs * ev0 + w4 * c0;
            const float m1 = w2 * a1 * es + w3 * xs * ev1 + w4 * c1;
            const float m2 = w2 * a2 * es + w3 * xs * ev2 + w4 * c2;

            float* ar = acc + (size_t)d * 64;
            afadd(ar + u, ms);
            afadd(ar + 16 + 3 * u + 0, m0);
            afadd(ar + 16 + 3 * u + 1, m1);
            afadd(ar + 16 + 3 * u + 2, m2);
        }
    }
}

// ---------------------------------------------------------------------------
// Kernel 2: gate node accumulators in place in d_out; scatter heavy nodes
// into segment sums (hs/hv/cnt) in workspace.
// ---------------------------------------------------------------------------
__global__ __launch_bounds__(256) void node_gate_kernel(
    const float* __restrict__ w_gate, const int* __restrict__ z,
    const int* __restrict__ canonical, float* __restrict__ out,
    float* __restrict__ hs, float* __restrict__ hv, float* __restrict__ cnt) {
    const int g = blockIdx.x * 256 + threadIdx.x;
    if (g >= N_NODES * 16) return;
    const int n = g >> 4;
    const int u = g & 15;

    float* row = out + (size_t)n * 64;
    const float ns = row[u];
    const float n0 = row[16 + 3 * u + 0];
    const float n1 = row[16 + 3 * u + 1];
    const float n2 = row[16 + 3 * u + 2];

    const float gs = w_gate[u] * ns;
    const float os = 1.0f / (1.0f + expf(-gs));
    const float th = tanhf(gs);
    const float wv = w_gate[16 + u];
    const float o0 = wv * n0 * th;
    const float o1 = wv * n1 * th;
    const float o2 = wv * n2 * th;

    row[u] = os;
    row[16 + 3 * u + 0] = o0;
    row[16 + 3 * u + 1] = o1;
    row[16 + 3 * u + 2] = o2;

    if (z[n] > 1) {
        const int c = canonical[n];
        afadd(hs + (size_t)c * 16 + u, os);
        float* hp = hv + ((size_t)c * 16 + u) * 3;
        afadd(hp + 0, o0);
        afadd(hp + 1, o1);
        afadd(hp + 2, o2);
        if (u == 0) afadd(cnt + c, 1.0f);
    }
}

// ---------------------------------------------------------------------------
// Kernel 3: heavy-segment mean + transform, in place on hs/hv.
// ---------------------------------------------------------------------------
__global__ __launch_bounds__(256) void heavy_tp_kernel(
    const float* __restrict__ w_heavy, const float* __restrict__ cnt,
    float* __restrict__ hs, float* __restrict__ hv) {
    const int g = blockIdx.x * 256 + threadIdx.x;
    if (g >= HSEG * 16) return;
    const int h = g >> 4;
    const int u = g & 15;

    const float dn = 1.0f / fmaxf(cnt[h], 1.0f);
    const float s = hs[(size_t)h * 16 + u] * dn;
    float* vp = hv + ((size_t)h * 16 + u) * 3;
    const float v0 = vp[0] * dn;
    const float v1 = vp[1] * dn;
    const float v2 = vp[2] * dn;

    const float ths = w_heavy[u] * s * s +
                      w_heavy[16 + u] * (v0 * v0 + v1 * v1 + v2 * v2);
    const float f = (w_heavy[32 + u] + w_heavy[48 + u]) * s;

    hs[(size_t)h * 16 + u] = ths;
    vp[0] = f * v0;
    vp[1] = f * v1;
    vp[2] = f * v2;
}

// ---------------------------------------------------------------------------
// Kernel 4: overwrite heavy rows of d_out with the transformed segment data.
// ---------------------------------------------------------------------------
__global__ __launch_bounds__(256) void heavy_select_kernel(
    const int* __restrict__ z, const int* __restrict__ canonical,
    const float* __restrict__ hs, const float* __restrict__ hv,
    float* __restrict__ out) {
    const int g = blockIdx.x * 256 + threadIdx.x;
    if (g >= N_NODES * 16) return;
    const int n = g >> 4;
    const int u = g & 15;
    if (z[n] <= 1) return;
    const int c = canonical[n];
    float* row = out + (size_t)n * 64;
    row[u] = hs[(size_t)c * 16 + u];
    const float* vp = hv + ((size_t)c * 16 + u) * 3;
    row[16 + 3 * u + 0] = vp[0];
    row[16 + 3 * u + 1] = vp[1];
    row[16 + 3 * u + 2] = vp[2];
}

extern "C" void kernel_launch(void* const* d_in, const int* in_sizes, int n_in,
                              void* d_out, int out_size, void* d_ws,
                              size_t ws_size, hipStream_t stream) {
    const float* x         = (const float*)d_in[0];
    const float* edge_attr = (const float*)d_in[1];
    const float* w_msg     = (const float*)d_in[2];
    const float* w_gate    = (const float*)d_in[3];
    const float* w_heavy   = (const float*)d_in[4];
    const int*   edge_idx  = (const int*)d_in[5];
    const int*   z         = (const int*)d_in[6];
    const int*   canonical = (const int*)d_in[7];
    float* out = (float*)d_out;

    float* hs  = (float*)d_ws;                  // HSEG*16
    float* hv  = hs + (size_t)HSEG * 16;        // HSEG*48
    float* cnt = hv + (size_t)HSEG * 48;        // HSEG

    (void)hipMemsetAsync(d_out, 0, (size_t)N_NODES * 64 * sizeof(float), stream);
    (void)hipMemsetAsync(d_ws, 0, (size_t)HSEG * 65 * sizeof(float), stream);

    edge_msg_kernel<<<N_EDGES / 512, 256, 0, stream>>>(x, edge_attr, w_msg,
                                                       edge_idx, out);
    node_gate_kernel<<<(N_NODES * 16 + 255) / 256, 256, 0, stream>>>(
        w_gate, z, canonical, out, hs, hv, cnt);
    heavy_tp_kernel<<<(HSEG * 16 + 255) / 256, 256, 0, stream>>>(w_heavy, cnt,
                                                                 hs, hv);
    heavy_select_kernel<<<(N_NODES * 16 + 255) / 256, 256, 0, stream>>>(
        z, canonical, hs, hv, out);
}